// MLSTMCell_68908455297201
// MI455X (gfx1250) — compile-verified
//
#include <hip/hip_runtime.h>
#include <math.h>

// ---------------------------------------------------------------------------
// mLSTM single step for MI455X (gfx1250, wave32).
//   gates = x @ W_i + b_i        -> i=exp, f=exp, o=sigmoid   (fp32 WMMA)
//   qkv   = blockdiag(x) @ W_qkv -> q, k/sqrt(D), v           (fp32 WMMA)
//   n_new = f*n + i*k ; denom = max(|n_new.q|,1)
//   h_new = o * (C q) / denom ; C_new = f*C + (i*v) k^T       (streaming)
// ---------------------------------------------------------------------------

typedef float v2f __attribute__((ext_vector_type(2)));
typedef float v8f __attribute__((ext_vector_type(8)));

#define BATCH 256
#define NHEAD 16
#define HD    128
#define HID   2048
#define NGATE 6144   // 3*HID
#define NQKV  384    // 3*HD

static __device__ __forceinline__ v8f wmma_f32(v2f a, v2f b, v8f c) {
  // D = A(16x4, f32) * B(4x16, f32) + C(16x16, f32)
  return __builtin_amdgcn_wmma_f32_16x16x4_f32(
      /*neg_a=*/false, a, /*neg_b=*/false, b,
      /*c_mod=*/(short)0, c, /*reuse_a=*/false, /*reuse_b=*/false);
}

// --------------------------- gate GEMM -------------------------------------
// x[B,HID] @ W_i[HID,NGATE] + b_i  -> exp/exp/sigmoid into ig/fg/og [B,HID]
__device__ __forceinline__ void gates_store(v8f acc, int row0, int col,
                                            const float* __restrict__ bias,
                                            float* __restrict__ ig,
                                            float* __restrict__ fg,
                                            float* __restrict__ og) {
  float bcol = bias[col];
#pragma unroll
  for (int rr = 0; rr < 8; ++rr) {
    int row = row0 + rr;
    float gv = acc[rr] + bcol;
    size_t base = (size_t)row * HID;
    if (col < HID)
      ig[base + col] = __expf(gv);
    else if (col < 2 * HID)
      fg[base + (col - HID)] = __expf(gv);
    else
      og[base + (col - 2 * HID)] = 1.0f / (1.0f + __expf(-gv));
  }
}

__global__ __launch_bounds__(256) void gates_kernel(
    const float* __restrict__ x, const float* __restrict__ W,
    const float* __restrict__ bias, float* __restrict__ ig,
    float* __restrict__ fg, float* __restrict__ og) {
  const int lane = threadIdx.x & 31;
  const int wave = threadIdx.x >> 5;
  const int wm = wave >> 2, wn = wave & 3;      // 2x4 waves -> 64x128 WG tile
  const int m_base = blockIdx.y * 64 + wm * 32; // wave tile 32x32
  const int n_base = blockIdx.x * 128 + wn * 32;
  const int r  = lane & 15;
  const int kq = (lane >> 4) << 1;              // 0 or 2

  // A fragments: lane holds A[row][kq..kq+1] (float2)
  const float* a0p = x + (size_t)(m_base + r) * HID + kq;
  const float* a1p = a0p + (size_t)16 * HID;
  // B fragments: lane holds B[kq..kq+1][col]
  const float* b0p = W + (size_t)kq * NGATE + (n_base + r);

  v8f acc00 = {}, acc01 = {}, acc10 = {}, acc11 = {};
  for (int k = 0; k < HID; k += 4) {
    v2f a0 = *(const v2f*)(a0p + k);
    v2f a1 = *(const v2f*)(a1p + k);
    const float* bk = b0p + (size_t)k * NGATE;
    v2f b0, b1;
    b0.x = bk[0];  b0.y = bk[NGATE];
    b1.x = bk[16]; b1.y = bk[16 + NGATE];
    acc00 = wmma_f32(a0, b0, acc00);
    acc01 = wmma_f32(a0, b1, acc01);
    acc10 = wmma_f32(a1, b0, acc10);
    acc11 = wmma_f32(a1, b1, acc11);
  }

  const int row_half = (lane >> 4) * 8;
  const int coll = lane & 15;
  gates_store(acc00, m_base +  0 + row_half, n_base +  0 + coll, bias, ig, fg, og);
  gates_store(acc01, m_base +  0 + row_half, n_base + 16 + coll, bias, ig, fg, og);
  gates_store(acc10, m_base + 16 + row_half, n_base +  0 + coll, bias, ig, fg, og);
  gates_store(acc11, m_base + 16 + row_half, n_base + 16 + coll, bias, ig, fg, og);
}

// --------------------------- QKV GEMM (per head) ---------------------------
// xh[B,HD] (strided out of x) @ W_qkv[h][HD,NQKV] + b_qkv[h] -> q,k,v [B,HID]
__device__ __forceinline__ void qkv_store(v8f acc, int row0, int j, int h,
                                          const float* __restrict__ bias,
                                          float* __restrict__ qo,
                                          float* __restrict__ ko,
                                          float* __restrict__ vo) {
  const float bcol = bias[h * NQKV + j];
  const float kscale = 0.08838834764831845f;  // 1/sqrt(128)
#pragma unroll
  for (int rr = 0; rr < 8; ++rr) {
    int row = row0 + rr;
    float val = acc[rr] + bcol;
    size_t base = (size_t)row * HID + h * HD;
    if (j < HD)
      qo[base + j] = val;
    else if (j < 2 * HD)
      ko[base + (j - HD)] = val * kscale;
    else
      vo[base + (j - 2 * HD)] = val;
  }
}

__global__ __launch_bounds__(256) void qkv_kernel(
    const float* __restrict__ x, const float* __restrict__ W,
    const float* __restrict__ bias, float* __restrict__ qo,
    float* __restrict__ ko, float* __restrict__ vo) {
  const int h = blockIdx.z;
  const int lane = threadIdx.x & 31;
  const int wave = threadIdx.x >> 5;
  const int wm = wave >> 2, wn = wave & 3;
  const int m_base = blockIdx.y * 64 + wm * 32;
  const int n_base = blockIdx.x * 128 + wn * 32;  // N = 384 -> 3 tiles
  const int r  = lane & 15;
  const int kq = (lane >> 4) << 1;

  const float* a0p = x + (size_t)(m_base + r) * HID + h * HD + kq;
  const float* a1p = a0p + (size_t)16 * HID;
  const float* Wh  = W + (size_t)h * HD * NQKV;
  const float* b0p = Wh + (size_t)kq * NQKV + (n_base + r);

  v8f acc00 = {}, acc01 = {}, acc10 = {}, acc11 = {};
  for (int k = 0; k < HD; k += 4) {
    v2f a0 = *(const v2f*)(a0p + k);
    v2f a1 = *(const v2f*)(a1p + k);
    const float* bk = b0p + (size_t)k * NQKV;
    v2f b0, b1;
    b0.x = bk[0];  b0.y = bk[NQKV];
    b1.x = bk[16]; b1.y = bk[16 + NQKV];
    acc00 = wmma_f32(a0, b0, acc00);
    acc01 = wmma_f32(a0, b1, acc01);
    acc10 = wmma_f32(a1, b0, acc10);
    acc11 = wmma_f32(a1, b1, acc11);
  }

  const int row_half = (lane >> 4) * 8;
  const int coll = lane & 15;
  qkv_store(acc00, m_base +  0 + row_half, n_base +  0 + coll, h, bias, qo, ko, vo);
  qkv_store(acc01, m_base +  0 + row_half, n_base + 16 + coll, h, bias, qo, ko, vo);
  qkv_store(acc10, m_base + 16 + row_half, n_base +  0 + coll, h, bias, qo, ko, vo);
  qkv_store(acc11, m_base + 16 + row_half, n_base + 16 + coll, h, bias, qo, ko, vo);
}

// --------------------------- state update ----------------------------------
// One block per (b,h). Streams C exactly once (536 MB total traffic => HBM
// bound, ~23 us floor at 23.3 TB/s). wave32 shuffle reductions for row dots.
__global__ __launch_bounds__(256) void update_kernel(
    const float* __restrict__ C, const float* __restrict__ n_in,
    const float* __restrict__ ig, const float* __restrict__ fg,
    const float* __restrict__ og, const float* __restrict__ q,
    const float* __restrict__ k, const float* __restrict__ v,
    float* __restrict__ h_out, float* __restrict__ C_out,
    float* __restrict__ n_out) {
  __shared__ __align__(16) float q_s[HD];
  __shared__ __align__(16) float k_s[HD];
  __shared__ float f_s[HD], iv_s[HD], o_s[HD];
  __shared__ float red[8];
  __shared__ float inv_denom_s;

  const int bh = blockIdx.x;  // b*NHEAD + h
  const int b = bh >> 4, h = bh & 15;
  const int t = threadIdx.x;
  const int lane = t & 31, wave = t >> 5;

  const size_t vecbase = (size_t)b * HID + h * HD;
  const float* Cb  = C + (size_t)bh * HD * HD;
  float* Cob       = C_out + (size_t)bh * HD * HD;

  // phase 1: n_new, stage per-head vectors, block-reduce n_new . q
  float partial = 0.0f;
  if (t < HD) {
    float iv = ig[vecbase + t];
    float fv = fg[vecbase + t];
    float ov = og[vecbase + t];
    float qv = q[vecbase + t];
    float kv = k[vecbase + t];
    float vv = v[vecbase + t];
    float nn = fv * n_in[vecbase + t] + iv * kv;
    n_out[vecbase + t] = nn;
    q_s[t] = qv; k_s[t] = kv; f_s[t] = fv; iv_s[t] = iv * vv; o_s[t] = ov;
    partial = nn * qv;
  }
  partial += __shfl_xor(partial, 16, 32);
  partial += __shfl_xor(partial,  8, 32);
  partial += __shfl_xor(partial,  4, 32);
  partial += __shfl_xor(partial,  2, 32);
  partial += __shfl_xor(partial,  1, 32);
  if (lane == 0) red[wave] = partial;
  __syncthreads();
  if (t == 0) {
    float dot = red[0] + red[1] + red[2] + red[3] +
                red[4] + red[5] + red[6] + red[7];
    float denom = fabsf(dot);
    denom = denom < 1.0f ? 1.0f : denom;
    inv_denom_s = 1.0f / denom;
  }
  __syncthreads();
  const float inv_denom = inv_denom_s;

  const float4 q4 = *(const float4*)(q_s + lane * 4);
  const float4 k4 = *(const float4*)(k_s + lane * 4);

  // phase 2: each wave owns 16 rows; one float4 per lane covers a 128-col row
  for (int it = 0; it < 16; ++it) {
    const int d = wave * 16 + it;
    const float4 c4 = *(const float4*)(Cb + (size_t)d * HD + lane * 4);
    const float fd = f_s[d], ivd = iv_s[d];
    float4 cn;
    cn.x = fd * c4.x + ivd * k4.x;
    cn.y = fd * c4.y + ivd * k4.y;
    cn.z = fd * c4.z + ivd * k4.z;
    cn.w = fd * c4.w + ivd * k4.w;
    *(float4*)(Cob + (size_t)d * HD + lane * 4) = cn;

    float p = c4.x * q4.x + c4.y * q4.y + c4.z * q4.z + c4.w * q4.w;
    p += __shfl_xor(p, 16, 32);
    p += __shfl_xor(p,  8, 32);
    p += __shfl_xor(p,  4, 32);
    p += __shfl_xor(p,  2, 32);
    p += __shfl_xor(p,  1, 32);
    if (lane == 0) h_out[vecbase + d] = o_s[d] * p * inv_denom;
  }
}

// --------------------------- launcher --------------------------------------
extern "C" void kernel_launch(void* const* d_in, const int* in_sizes, int n_in,
                              void* d_out, int out_size, void* d_ws,
                              size_t ws_size, hipStream_t stream) {
  const float* x     = (const float*)d_in[0];
  // d_in[1] = h (unused by the reference recurrence)
  const float* C     = (const float*)d_in[2];
  const float* n     = (const float*)d_in[3];
  const float* W_i   = (const float*)d_in[4];
  const float* b_i   = (const float*)d_in[5];
  const float* W_qkv = (const float*)d_in[6];
  const float* b_qkv = (const float*)d_in[7];

  float* ws = (float*)d_ws;
  const size_t P = (size_t)BATCH * HID;  // 524288 floats per plane
  float* ig = ws + 0 * P;
  float* fg = ws + 1 * P;
  float* og = ws + 2 * P;
  float* qb = ws + 3 * P;
  float* kb = ws + 4 * P;
  float* vb = ws + 5 * P;

  float* out   = (float*)d_out;
  float* h_out = out;                                   // [B, HID]
  float* C_out = out + P;                               // [B, NH, D, D]
  float* n_out = C_out + (size_t)BATCH * NHEAD * HD * HD;  // [B, NH, D]

  dim3 g1(NGATE / 128, BATCH / 64, 1);   // 48 x 4
  gates_kernel<<<g1, 256, 0, stream>>>(x, W_i, b_i, ig, fg, og);

  dim3 g2(NQKV / 128, BATCH / 64, NHEAD);  // 3 x 4 x 16
  qkv_kernel<<<g2, 256, 0, stream>>>(x, W_qkv, b_qkv, qb, kb, vb);

  update_kernel<<<BATCH * NHEAD, 256, 0, stream>>>(
      C, n, ig, fg, og, qb, kb, vb, h_out, C_out, n_out);
}